// DeepGraphSAGE_24163486007932
// MI455X (gfx1250) — compile-verified
//
#include <hip/hip_runtime.h>

#define N_NODES 50000
#define NE      800000
#define D       128
#define EPS_BN  1e-5f
#define MTILES  (N_NODES / 16)   // 3125, exact
#define LDSS    132              // LDS row stride (floats), conflict-free for WMMA A reads

typedef __attribute__((ext_vector_type(2))) float v2f;
typedef __attribute__((ext_vector_type(4))) float v4f;
typedef __attribute__((ext_vector_type(8))) float v8f;

// ---------------- utility kernels ----------------

__global__ void zero_kernel(float* __restrict__ p, int n4) {
    int i = blockIdx.x * blockDim.x + threadIdx.x;
    if (i < n4) ((v4f*)p)[i] = (v4f){0.0f, 0.0f, 0.0f, 0.0f};
}

__global__ void degree_kernel(const int* __restrict__ dst, float* __restrict__ deg) {
    int e = blockIdx.x * blockDim.x + threadIdx.x;
    if (e < NE) atomicAdd(&deg[dst[e]], 1.0f);
}

__global__ void invdeg_kernel(float* __restrict__ deg) {
    int i = blockIdx.x * blockDim.x + threadIdx.x;
    if (i < N_NODES) {
        float d = deg[i];
        deg[i] = 1.0f / fmaxf(d, 1.0f);
    }
}

// gather x[src] and scatter-add into agg[dst]; 128 threads (one per column) per edge
__global__ void scatter_kernel(const float* __restrict__ x, const int* __restrict__ src,
                               const int* __restrict__ dst, float* __restrict__ agg) {
    int t = threadIdx.x;
    int e = blockIdx.x * 2 + (t >> 7);
    int d = t & 127;
    if (e < NE) {
        int s  = src[e];
        int dd = dst[e];
        atomicAdd(&agg[(size_t)dd * D + d], x[(size_t)s * D + d]);
    }
}

// ---------------- fused WMMA GEMM + epilogue ----------------
// mode 0: out = relu(Ax @ W1 + bias)                                (projection)
// mode 1: h = (Aagg*invdeg) @ W1 + bias + Ax @ W2; out = Ax + relu(h*s + t)   (mid layers)
// mode 2: same as 1 but no residual                                 (last layer)
__global__ __launch_bounds__(256, 1)
void sage_gemm_kernel(const float* __restrict__ Aagg, const float* __restrict__ inv_deg,
                      const float* __restrict__ Ax,
                      const float* __restrict__ W1, const float* __restrict__ W2,
                      const float* __restrict__ bias,
                      const float* __restrict__ gamma, const float* __restrict__ beta,
                      const float* __restrict__ mean, const float* __restrict__ var,
                      float* __restrict__ out, int mode) {
    __shared__ float sA1[16 * LDSS];
    __shared__ float sA2[16 * LDSS];

    const int t    = threadIdx.x;
    const int lane = t & 31;
    const int wave = t >> 5;        // 0..7 -> output N-tile
    const int half = lane >> 4;     // 0/1: which K-pair this lane carries
    const int lrow = lane & 15;
    const int n    = wave * 16 + lrow;   // output column owned by this lane

    // B fragments in registers, reused for every M-tile.
    // B 4x16 layout: VGPR0 = row K=k0+2*half, VGPR1 = row K=k0+2*half+1, cols across lanes.
    v2f b1[32];
    v2f b2[32];
#pragma unroll
    for (int ks = 0; ks < 32; ++ks) {
        int k = ks * 4 + 2 * half;
        b1[ks].x = W1[k * D + n];
        b1[ks].y = W1[(k + 1) * D + n];
    }
    if (mode != 0) {
#pragma unroll
        for (int ks = 0; ks < 32; ++ks) {
            int k = ks * 4 + 2 * half;
            b2[ks].x = W2[k * D + n];
            b2[ks].y = W2[(k + 1) * D + n];
        }
    }

    // per-column epilogue constants: y = h*s + tt (BN folded), bias
    float bcol = bias[n];
    float s = 1.0f, tt = 0.0f;
    if (mode != 0) {
        s  = gamma[n] * rsqrtf(var[n] + EPS_BN);
        tt = beta[n] - mean[n] * s;
    }

    for (int mt = blockIdx.x; mt < MTILES; mt += gridDim.x) {
        const size_t rowbase = (size_t)mt * 16;

        // cooperative load: 16x128 tile(s) -> LDS (8 floats per thread, coalesced)
#pragma unroll
        for (int i = 0; i < 8; ++i) {
            int idx = t + 256 * i;       // 0..2047
            int r   = idx >> 7;          // 0..15
            int c   = idx & 127;
            float xv = Ax[(rowbase + r) * D + c];
            if (mode == 0) {
                sA1[r * LDSS + c] = xv;
            } else {
                sA1[r * LDSS + c] = Aagg[(rowbase + r) * D + c] * inv_deg[rowbase + r];
                sA2[r * LDSS + c] = xv;
            }
        }
        __syncthreads();

        v8f c8 = {};
        // GEMM 1: A1 (16x128) x W1 (128x16) via 32 x WMMA f32 16x16x4
#pragma unroll
        for (int ks = 0; ks < 32; ++ks) {
            int k = ks * 4 + 2 * half;
            v2f a = *(const v2f*)&sA1[lrow * LDSS + k];
            c8 = __builtin_amdgcn_wmma_f32_16x16x4_f32(false, a, false, b1[ks],
                                                       (short)0, c8, false, false);
        }
        if (mode != 0) {
            // GEMM 2: x @ W2 accumulated into the same tile
#pragma unroll
            for (int ks = 0; ks < 32; ++ks) {
                int k = ks * 4 + 2 * half;
                v2f a = *(const v2f*)&sA2[lrow * LDSS + k];
                c8 = __builtin_amdgcn_wmma_f32_16x16x4_f32(false, a, false, b2[ks],
                                                           (short)0, c8, false, false);
            }
        }

        // epilogue: lane holds rows m = v + 8*half, column n
#pragma unroll
        for (int v = 0; v < 8; ++v) {
            int m   = v + 8 * half;
            float h = c8[v] + bcol;
            float y;
            if (mode == 0) {
                y = fmaxf(h, 0.0f);
            } else {
                y = fmaxf(h * s + tt, 0.0f);
                if (mode == 1) y += sA2[m * LDSS + n];   // residual from x tile in LDS
            }
            out[(rowbase + m) * D + n] = y;
        }
        __syncthreads();
    }
}

// ---------------- host launcher ----------------

extern "C" void kernel_launch(void* const* d_in, const int* in_sizes, int n_in,
                              void* d_out, int out_size, void* d_ws, size_t ws_size,
                              hipStream_t stream) {
    (void)in_sizes; (void)n_in; (void)out_size; (void)ws_size;

    const float* x   = (const float*)d_in[0];
    const int*   ei  = (const int*)d_in[1];
    const float* Wp  = (const float*)d_in[2];
    const float* bp  = (const float*)d_in[3];
    const float* Wl  = (const float*)d_in[4];
    const float* bl  = (const float*)d_in[5];
    const float* Wr  = (const float*)d_in[6];
    const float* gam = (const float*)d_in[7];
    const float* bet = (const float*)d_in[8];
    const float* mea = (const float*)d_in[9];
    const float* var = (const float*)d_in[10];

    const int* src = ei;
    const int* dst = ei + NE;

    float* ws  = (float*)d_ws;
    float* deg = ws;                                   // N floats (becomes inv_deg)
    float* xa  = ws + ((N_NODES + 63) / 64) * 64;      // N*D
    float* xb  = xa + (size_t)N_NODES * D;             // N*D
    float* agg = xb + (size_t)N_NODES * D;             // N*D

    const int GEMM_BLOCKS = 512;

    // degree -> inv_deg (computed once, reused by all layers)
    {
        int n4 = N_NODES / 4;  // 12500, exact
        zero_kernel<<<(n4 + 255) / 256, 256, 0, stream>>>(deg, n4);
        degree_kernel<<<(NE + 255) / 256, 256, 0, stream>>>(dst, deg);
        invdeg_kernel<<<(N_NODES + 255) / 256, 256, 0, stream>>>(deg);
    }

    // projection: xa = relu(x @ Wp + bp)
    sage_gemm_kernel<<<GEMM_BLOCKS, 256, 0, stream>>>(
        nullptr, deg, x, Wp, Wp, bp, gam, bet, mea, var, xa, 0);

    float* xcur  = xa;
    float* xnext = xb;
    for (int i = 0; i < 4; ++i) {
        int n4 = (N_NODES * D) / 4;  // 1.6M float4
        zero_kernel<<<(n4 + 255) / 256, 256, 0, stream>>>(agg, n4);
        scatter_kernel<<<NE / 2, 256, 0, stream>>>(xcur, src, dst, agg);

        float* o = (i < 3) ? xnext : (float*)d_out;
        sage_gemm_kernel<<<GEMM_BLOCKS, 256, 0, stream>>>(
            agg, deg, xcur,
            Wl + (size_t)i * D * D, Wr + (size_t)i * D * D, bl + (size_t)i * D,
            gam + (size_t)i * D, bet + (size_t)i * D, mea + (size_t)i * D, var + (size_t)i * D,
            o, (i < 3) ? 1 : 2);

        float* tmp = xcur; xcur = xnext; xnext = tmp;
    }
}